// GMMSA_64802466562881
// MI455X (gfx1250) — compile-verified
//
#include <hip/hip_runtime.h>

typedef __attribute__((ext_vector_type(2))) float v2f;
typedef __attribute__((ext_vector_type(8))) float v8f;

#define IMG 192
#define HW  36864   // 192*192

// D = A(16x4) * B(4x16) + C, fp32 WMMA (CDNA5)
__device__ __forceinline__ v8f wmma4(v2f a, v2f b, v8f c) {
  return __builtin_amdgcn_wmma_f32_16x16x4_f32(false, a, false, b, (short)0, c,
                                               false, false);
}

// ---------------------------------------------------------------------------
// 1x1-conv GEMM:  Y[z][m][n] = sum_k W[m][k] * X[z][k][n] + bias[m]
// K = 180 fixed. Block: 256 thr (8 waves), tile N=32 staged in LDS.
// LDS B-tile is k-pair interleaved: (k,n) -> xs[(k>>1)*96 + 2n + (k&1)]
// so each B fragment is a single aligned ds_load_b64; pair-row stride 96
// floats == 32 banks, so half-waves (koff 0 vs 2) use disjoint bank halves.
// ---------------------------------------------------------------------------
template<int M>
__global__ __launch_bounds__(256) void gemm1x1(const float* __restrict__ W,
                                               const float* __restrict__ X,
                                               const float* __restrict__ bias,
                                               float* __restrict__ Y) {
  constexpr int MT = (M + 15) / 16;     // 23 (proj) or 12 (fuse)
  __shared__ __align__(16) float xs[90 * 96];   // 34.5 KB
  const int tid  = threadIdx.x;
  const int lane = tid & 31, wave = tid >> 5;
  const int lo16 = lane & 15, hi = lane >> 4;
  const int koff = hi * 2, hi8 = hi * 8;
  const int z  = blockIdx.z;
  const int n0 = blockIdx.x * 32;
  const float* Xb = X + (size_t)z * 180 * HW;

  for (int idx = tid; idx < 180 * 32; idx += 256) {
    int k = idx >> 5, n = idx & 31;
    xs[(k >> 1) * 96 + 2 * n + (k & 1)] = Xb[(size_t)k * HW + n0 + n];
  }
  __syncthreads();

  for (int t = wave; t < MT; t += 8) {           // wave-uniform trip count
    const int rowA = t * 16 + lo16;
    const int rowL = (rowA < M) ? rowA : (M - 1);  // clamp: always in-bounds
    const float* Wr = W + (size_t)rowL * 180 + koff; // even index -> 8B aligned
    v8f acc0 = {}, acc1 = {};
    #pragma unroll 5
    for (int ks = 0; ks < 45; ++ks) {
      v2f a  = *(const v2f*)(Wr + ks * 4);                    // global b64
      const float* bp = &xs[(ks * 2 + hi) * 96 + 2 * lo16];
      v2f b0 = *(const v2f*)(bp);                             // ds_load_b64
      v2f b1 = *(const v2f*)(bp + 32);                        // ds_load_b64
      acc0 = wmma4(a, b0, acc0);
      acc1 = wmma4(a, b1, acc1);
    }
    if (t * 16 + 16 <= M) {        // wave-uniform fast path: no guards
      #pragma unroll
      for (int r = 0; r < 8; ++r) {
        int row = t * 16 + r + hi8;
        float bs = bias[row];
        size_t base = ((size_t)z * M + row) * HW + n0;
        Y[base + lo16]      = acc0[r] + bs;
        Y[base + 16 + lo16] = acc1[r] + bs;
      }
    } else {                       // last partial tile only
      #pragma unroll
      for (int r = 0; r < 8; ++r) {
        int row = t * 16 + r + hi8;
        if (row < M) {
          float bs = bias[row];
          size_t base = ((size_t)z * M + row) * HW + n0;
          Y[base + lo16]      = acc0[r] + bs;
          Y[base + 16 + lo16] = acc1[r] + bs;
        }
      }
    }
  }
}

// ---------------------------------------------------------------------------
// GroupNorm statistics over xp[2][360][HW], 2 groups of 180 channels.
// ---------------------------------------------------------------------------
__global__ __launch_bounds__(64) void gn_init(float* stats) {
  if (threadIdx.x < 8) stats[threadIdx.x] = 0.f;
}

__global__ __launch_bounds__(256) void gn_stats(const float* __restrict__ xp,
                                                float* __restrict__ stats) {
  const int bg = blockIdx.y;                // 0..3  (b*2+g)
  const int b = bg >> 1, g = bg & 1;
  const float* p = xp + ((size_t)b * 360 + g * 180) * HW;
  const size_t N = (size_t)180 * HW;
  float s = 0.f, s2 = 0.f;
  for (size_t i = (size_t)blockIdx.x * 256 + threadIdx.x; i < N;
       i += (size_t)gridDim.x * 256) {
    float v = p[i];
    s += v; s2 += v * v;
  }
  __shared__ float rs[256], rq[256];
  int tid = threadIdx.x;
  rs[tid] = s; rq[tid] = s2;
  __syncthreads();
  for (int o = 128; o > 0; o >>= 1) {
    if (tid < o) { rs[tid] += rs[tid + o]; rq[tid] += rq[tid + o]; }
    __syncthreads();
  }
  if (tid == 0) {
    atomicAdd(&stats[bg], rs[0]);
    atomicAdd(&stats[4 + bg], rq[0]);
  }
}

__global__ __launch_bounds__(64) void gn_final(const float* __restrict__ stats,
                                               float* __restrict__ mr) {
  int i = threadIdx.x;
  if (i < 4) {
    const float N = 180.f * (float)HW;
    float m   = stats[i] / N;
    float var = stats[4 + i] / N - m * m;
    mr[i]     = m;
    mr[4 + i] = rsqrtf(var + 1e-5f);
  }
}

// ---------------------------------------------------------------------------
// Window attention for one scale. One block = (b, wy, wx, head), 128 thr.
// q/v loaded from xp with roll + GroupNorm folded in; S = q_l q_l^T via WMMA;
// row softmax; attn written to d_out; O = P V via WMMA, scaled into y_fused.
// ---------------------------------------------------------------------------
template<int SCALE, int WS, int HE, int C, int CP, int TM>
__global__ __launch_bounds__(128) void attn_kernel(
    const float* __restrict__ xp, const float* __restrict__ mr,
    const float* __restrict__ gamma, const float* __restrict__ beta,
    const float* __restrict__ swv, float* __restrict__ attn_out,
    float* __restrict__ yfused) {
  constexpr int T  = WS * WS;
  constexpr int NH = IMG / WS;
  constexpr int QS = CP + 1;     // padded LDS stride for q/v
  constexpr int SS = T + 1;      // padded LDS stride for S
  __shared__ float q_s[T * QS];
  __shared__ float v_s[T * QS];
  __shared__ float S_s[TM * SS];

  const int tid  = threadIdx.x;
  const int lane = tid & 31, wave = tid >> 5;
  const int lo16 = lane & 15, koff = (lane >> 4) * 2, hi8 = (lane >> 4) * 8;

  const int bi   = blockIdx.x;
  const int head = bi % HE;
  const int w    = bi / HE;
  const int wx   = w % NH;
  const int wy   = (w / NH) % NH;
  const int b    = w / (NH * NH);

  const float qscale = rsqrtf((float)HE);
  const float sw     = swv[SCALE];

  // zero the padded tail columns
  for (int idx = tid; idx < T * (CP - C); idx += 128) {
    int t = idx / (CP - C), c = C + idx % (CP - C);
    q_s[t * QS + c] = 0.f;
    v_s[t * QS + c] = 0.f;
  }
  // load q/v with spatial roll and fused GroupNorm
  for (int idx = tid; idx < T * C; idx += 128) {
    int t = idx / C, c = idx % C;
    int ty = t / WS, tx = t % WS;
    int hg = (wy * WS + ty - WS / 2 + IMG) % IMG;
    int wg = (wx * WS + tx + WS / 2) % IMG;
    size_t pix = (size_t)hg * IMG + wg;
    int qch = SCALE * 120 + c * HE + head;
    int vch = qch + 60;
    float xq = xp[((size_t)b * 360 + qch) * HW + pix];
    float xv = xp[((size_t)b * 360 + vch) * HW + pix];
    int gq = (qch >= 180) ? 1 : 0, gv = (vch >= 180) ? 1 : 0;
    float mq = mr[b * 2 + gq], rq = mr[4 + b * 2 + gq];
    float mv = mr[b * 2 + gv], rv = mr[4 + b * 2 + gv];
    q_s[t * QS + c] = ((xq - mq) * rq * gamma[qch] + beta[qch]) * qscale;
    v_s[t * QS + c] =  (xv - mv) * rv * gamma[vch] + beta[vch];
  }
  __syncthreads();

  float* aout = attn_out + ((size_t)w * HE + head) * T * T;

  for (int m0 = 0; m0 < T; m0 += TM) {
    // ---- S = q_l q_l^T, row block [m0, m0+TM) x [0, T) ----
    constexpr int NTN = T / 16, NTM = TM / 16;
    for (int tile = wave; tile < NTM * NTN; tile += 4) {
      int tm = (tile / NTN) * 16, tn = (tile % NTN) * 16;
      int am = m0 + tm + lo16;
      int bn = tn + lo16;
      v8f acc = {};
      #pragma unroll
      for (int k = 0; k < CP; k += 4) {
        v2f a, bb;
        a.x  = q_s[am * QS + k + koff];
        a.y  = q_s[am * QS + k + koff + 1];
        bb.x = q_s[bn * QS + k + koff];
        bb.y = q_s[bn * QS + k + koff + 1];
        acc = wmma4(a, bb, acc);
      }
      #pragma unroll
      for (int r = 0; r < 8; ++r)
        S_s[(tm + r + hi8) * SS + tn + lo16] = acc[r];
    }
    __syncthreads();

    // ---- row softmax (one wave per row), write attn to d_out ----
    for (int row = wave; row < TM; row += 4) {
      float* Sr = &S_s[row * SS];
      float m = -3.4e38f;
      for (int t = lane; t < T; t += 32) m = fmaxf(m, Sr[t]);
      for (int o = 16; o > 0; o >>= 1) m = fmaxf(m, __shfl_xor(m, o, 32));
      float s = 0.f;
      for (int t = lane; t < T; t += 32) {
        float e = expf(Sr[t] - m);
        Sr[t] = e; s += e;
      }
      for (int o = 16; o > 0; o >>= 1) s += __shfl_xor(s, o, 32);
      float inv = 1.f / s;
      float* ar = aout + (size_t)(m0 + row) * T;
      for (int t = lane; t < T; t += 32) {
        float p = Sr[t] * inv;
        Sr[t] = p;
        ar[t] = p;
      }
    }
    __syncthreads();

    // ---- O = P * V for this row block ----
    constexpr int NON = CP / 16;
    for (int tile = wave; tile < NTM * NON; tile += 4) {
      int tm = (tile / NON) * 16, tn = (tile % NON) * 16;
      v8f acc = {};
      for (int k = 0; k < T; k += 4) {
        v2f a, bb;
        a.x  = S_s[(tm + lo16) * SS + k + koff];
        a.y  = S_s[(tm + lo16) * SS + k + koff + 1];
        bb.x = v_s[(k + koff) * QS + tn + lo16];
        bb.y = v_s[(k + koff + 1) * QS + tn + lo16];
        acc = wmma4(a, bb, acc);
      }
      int c = tn + lo16;
      if (c < C) {
        int och = SCALE * 60 + c * HE + head;
        #pragma unroll
        for (int r = 0; r < 8; ++r) {
          int t  = m0 + tm + r + hi8;
          int ty = t / WS, tx = t % WS;
          int hg = (wy * WS + ty - WS / 2 + IMG) % IMG;
          int wg = (wx * WS + tx + WS / 2) % IMG;
          yfused[((size_t)b * 180 + och) * HW + (size_t)hg * IMG + wg] =
              sw * acc[r];
        }
      }
    }
    __syncthreads();
  }
}

// ---------------------------------------------------------------------------
extern "C" void kernel_launch(void* const* d_in, const int* in_sizes, int n_in,
                              void* d_out, int out_size, void* d_ws,
                              size_t ws_size, hipStream_t stream) {
  (void)in_sizes; (void)n_in; (void)out_size; (void)ws_size;
  const float* x        = (const float*)d_in[0];   // (2,180,192,192)
  const float* proj_w   = (const float*)d_in[1];   // (360,180)
  const float* proj_b   = (const float*)d_in[2];   // (360)
  const float* gn_gamma = (const float*)d_in[3];   // (360)
  const float* gn_beta  = (const float*)d_in[4];   // (360)
  // d_in[5] = token_scores : global branch is numerically identity -> unused
  const float* scale_w  = (const float*)d_in[6];   // (3)
  const float* fuse_w   = (const float*)d_in[7];   // (180,180)
  const float* fuse_b   = (const float*)d_in[8];   // (180)
  float* out = (float*)d_out;

  // workspace layout
  float* xp     = (float*)d_ws;                    // 2*360*HW
  float* yfused = xp + (size_t)2 * 360 * HW;       // 2*180*HW
  float* stats  = yfused + (size_t)2 * 180 * HW;   // 8 floats (sum,sumsq)
  float* mrbuf  = stats + 8;                       // 8 floats (mean,rstd)

  // output layout: y, atn0, atn1, atn2
  float* y_out = out;
  float* a0 = out + (size_t)2 * 180 * HW;              // 13,271,040
  float* a1 = a0 + (size_t)2 * 48 * 48 * 2 * 16 * 16;  // + 2,359,296
  float* a2 = a1 + (size_t)1152 * 4 * 64 * 64;         // +18,874,368

  // 1) projection GEMM -> xp
  gemm1x1<360><<<dim3(HW / 32, 1, 2), 256, 0, stream>>>(proj_w, x, proj_b, xp);

  // 2) GroupNorm stats
  gn_init<<<1, 64, 0, stream>>>(stats);
  gn_stats<<<dim3(864, 4), 256, 0, stream>>>(xp, stats);
  gn_final<<<1, 64, 0, stream>>>(stats, mrbuf);

  // 3) window attention per scale (GN folded into loads)
  attn_kernel<0, 4, 2, 30, 32, 16>
      <<<2 * 48 * 48 * 2, 128, 0, stream>>>(xp, mrbuf, gn_gamma, gn_beta,
                                            scale_w, a0, yfused);
  attn_kernel<1, 8, 4, 15, 16, 64>
      <<<2 * 24 * 24 * 4, 128, 0, stream>>>(xp, mrbuf, gn_gamma, gn_beta,
                                            scale_w, a1, yfused);
  attn_kernel<2, 12, 6, 10, 16, 48>
      <<<2 * 16 * 16 * 6, 128, 0, stream>>>(xp, mrbuf, gn_gamma, gn_beta,
                                            scale_w, a2, yfused);

  // 4) fuse GEMM -> y
  gemm1x1<180><<<dim3(HW / 32, 1, 2), 256, 0, stream>>>(fuse_w, yfused, fuse_b,
                                                        y_out);
}